// HeteroGNNEncoder_80865644249440
// MI455X (gfx1250) — compile-verified
//
#include <hip/hip_runtime.h>

// ---------------------------------------------------------------------------
// GraphSAGE encoder for MI455X (gfx1250, wave32).
// Bandwidth-bound workload (~11.5 GFLOP vs ~300MB traffic @ 23.3 TB/s);
// intermediates (~77MB) are L2-resident (192MB L2), so we stage through d_ws.
// GEMMs use V_WMMA_F32_16X16X4_F32 (native f32 matrix pipe) with the LayerNorm
// + ReLU epilogue fused into the GEMM via wave32 shfl_xor row reductions.
// ---------------------------------------------------------------------------

#define N_NODES 50000
#define N_EDGES 640000
#define D 128              // hidden dim (= D_IN = D_H)
#define D_EMB 256
#define OUT_STRIDE (D + D_EMB)   // 384
#define N_ROWTILES (N_NODES / 16)  // 3125 exactly

typedef float v2f __attribute__((ext_vector_type(2)));
typedef float v8f __attribute__((ext_vector_type(8)));

// D = A(16x4 f32) * B(4x16 f32) + C(16x16 f32), one matrix per wave32.
__device__ __forceinline__ v8f wmma_f32(v2f a, v2f b, v8f c) {
  // 8 args: (neg_a, A, neg_b, B, c_mod, C, reuse_a, reuse_b)
  return __builtin_amdgcn_wmma_f32_16x16x4_f32(
      false, a, false, b, (short)0, c, false, false);
}

// ---------------------------------------------------------------------------
// Zero scratch (agg sums + counts are contiguous, one launch covers both)
// ---------------------------------------------------------------------------
__global__ void k_zero(float* __restrict__ p, int n) {
  int i = blockIdx.x * blockDim.x + threadIdx.x;
  if (i < n) p[i] = 0.0f;
}

// ---------------------------------------------------------------------------
// Copy x into out[:, 0:128] (row stride 384)
// ---------------------------------------------------------------------------
__global__ void k_copy_x(const float* __restrict__ x, float* __restrict__ out) {
  int i = blockIdx.x * blockDim.x + threadIdx.x;     // N*32 workers, float4 each
  if (i >= N_NODES * 32) return;
  int r = i >> 5;
  int c = (i & 31) * 4;
  const float4 v = *(const float4*)(x + (size_t)r * D + c);
  *(float4*)(out + (size_t)r * OUT_STRIDE + c) = v;
}

// ---------------------------------------------------------------------------
// Mean-aggregation scatter: one wave per edge.
// Lane loads float4 of source row (coalesced 512B per wave), hardware f32
// atomics into the L2-resident sum buffer; lane 0 bumps the degree count.
// ---------------------------------------------------------------------------
__global__ __launch_bounds__(256) void k_aggregate(
    const int* __restrict__ src, const int* __restrict__ dst,
    const float* __restrict__ h, float* __restrict__ aggS,
    float* __restrict__ cnt) {
  const int lane = threadIdx.x & 31;
  const int e = blockIdx.x * 8 + (threadIdx.x >> 5);
  if (e >= N_EDGES) return;
  const int s = src[e];
  const int d = dst[e];
  const float4 m = *(const float4*)(h + (size_t)s * D + lane * 4);
  float* ap = aggS + (size_t)d * D + lane * 4;
  __hip_atomic_fetch_add(ap + 0, m.x, __ATOMIC_RELAXED, __HIP_MEMORY_SCOPE_AGENT);
  __hip_atomic_fetch_add(ap + 1, m.y, __ATOMIC_RELAXED, __HIP_MEMORY_SCOPE_AGENT);
  __hip_atomic_fetch_add(ap + 2, m.z, __ATOMIC_RELAXED, __HIP_MEMORY_SCOPE_AGENT);
  __hip_atomic_fetch_add(ap + 3, m.w, __ATOMIC_RELAXED, __HIP_MEMORY_SCOPE_AGENT);
  if (lane == 0)
    __hip_atomic_fetch_add(cnt + d, 1.0f, __ATOMIC_RELAXED, __HIP_MEMORY_SCOPE_AGENT);
}

// ---------------------------------------------------------------------------
// Shared fused epilogue: bias + LayerNorm + ReLU + store.
// C layout: lane holds element (m = v + 8*(lane>=16), n = t*16 + (lane&15)).
// Row stats reduce over the 16 lanes of each half with shfl_xor 1/2/4/8.
// ---------------------------------------------------------------------------
__device__ __forceinline__ void ln_relu_store(
    v8f acc[8], const float* __restrict__ bias, const float* __restrict__ gamma,
    const float* __restrict__ beta, float* __restrict__ out,
    int row0, int lm, int half) {
  float bn[8], gn[8], bt[8];
#pragma unroll
  for (int t = 0; t < 8; ++t) {
    const int n = t * 16 + lm;
    bn[t] = bias[n];
    gn[t] = gamma[n];
    bt[t] = beta[n];
  }
#pragma unroll
  for (int t = 0; t < 8; ++t)
#pragma unroll
    for (int v = 0; v < 8; ++v) acc[t][v] += bn[t];

#pragma unroll
  for (int v = 0; v < 8; ++v) {
    float s = 0.0f, sq = 0.0f;
#pragma unroll
    for (int t = 0; t < 8; ++t) {
      const float c = acc[t][v];
      s += c;
      sq += c * c;
    }
#pragma unroll
    for (int m = 1; m < 16; m <<= 1) {   // stays inside each 16-lane half
      s += __shfl_xor(s, m, 32);
      sq += __shfl_xor(sq, m, 32);
    }
    const float mu = s * (1.0f / (float)D);
    const float var = sq * (1.0f / (float)D) - mu * mu;
    const float rs = rsqrtf(var + 1e-5f);
    const int row = row0 + v + half * 8;
#pragma unroll
    for (int t = 0; t < 8; ++t) {
      const float val = (acc[t][v] - mu) * rs * gn[t] + bt[t];
      out[(size_t)row * D + t * 16 + lm] = fmaxf(val, 0.0f);
    }
  }
}

// ---------------------------------------------------------------------------
// pre layer: out = ReLU(LN(in @ W^T + b)).  One wave = 16 rows x 128 cols.
// ---------------------------------------------------------------------------
__global__ __launch_bounds__(256) void k_lin_ln_relu(
    const float* __restrict__ in, const float* __restrict__ W,
    const float* __restrict__ bias, const float* __restrict__ gamma,
    const float* __restrict__ beta, float* __restrict__ out) {
  const int lane = threadIdx.x & 31;
  const int rowTile = blockIdx.x * 8 + (threadIdx.x >> 5);
  if (rowTile >= N_ROWTILES) return;
  const int row0 = rowTile * 16;
  const int lm = lane & 15;
  const int half = lane >> 4;

  const float* arow = in + (size_t)(row0 + lm) * D + half * 2;
  v8f acc[8] = {};
  for (int k = 0; k < D; k += 4) {
    const v2f a = *(const v2f*)(arow + k);
#pragma unroll
    for (int t = 0; t < 8; ++t) {
      const int n = t * 16 + lm;
      const v2f b = *(const v2f*)(W + (size_t)n * D + k + half * 2);
      acc[t] = wmma_f32(a, b, acc[t]);
    }
  }
  ln_relu_store(acc, bias, gamma, beta, out, row0, lm, half);
}

// ---------------------------------------------------------------------------
// SAGE layer: out = ReLU(LN((agg/cnt) @ Wl^T + h @ Wr^T + b)).
// Mean normalization fused into the A-load (per-row 1/max(cnt,1)).
// ---------------------------------------------------------------------------
__global__ __launch_bounds__(256) void k_sage_ln_relu(
    const float* __restrict__ aggS, const float* __restrict__ cnt,
    const float* __restrict__ h, const float* __restrict__ Wl,
    const float* __restrict__ Wr, const float* __restrict__ bias,
    const float* __restrict__ gamma, const float* __restrict__ beta,
    float* __restrict__ out) {
  const int lane = threadIdx.x & 31;
  const int rowTile = blockIdx.x * 8 + (threadIdx.x >> 5);
  if (rowTile >= N_ROWTILES) return;
  const int row0 = rowTile * 16;
  const int lm = lane & 15;
  const int half = lane >> 4;

  const float rinv = 1.0f / fmaxf(cnt[row0 + lm], 1.0f);  // A rows: m = lane&15
  const float* srow = aggS + (size_t)(row0 + lm) * D + half * 2;
  const float* hrow = h + (size_t)(row0 + lm) * D + half * 2;

  v8f acc[8] = {};
  for (int k = 0; k < D; k += 4) {
    v2f aa = *(const v2f*)(srow + k);
    aa.x *= rinv;
    aa.y *= rinv;
    const v2f ah = *(const v2f*)(hrow + k);
#pragma unroll
    for (int t = 0; t < 8; ++t) {
      const int n = t * 16 + lm;
      const v2f bl = *(const v2f*)(Wl + (size_t)n * D + k + half * 2);
      acc[t] = wmma_f32(aa, bl, acc[t]);
      const v2f br = *(const v2f*)(Wr + (size_t)n * D + k + half * 2);
      acc[t] = wmma_f32(ah, br, acc[t]);
    }
  }
  ln_relu_store(acc, bias, gamma, beta, out, row0, lm, half);
}

// ---------------------------------------------------------------------------
// Final linear: out[:, 128:384] = h @ lin_w^T + lin_b.  Wave per 16x16 tile.
// ---------------------------------------------------------------------------
__global__ __launch_bounds__(256) void k_final_gemm(
    const float* __restrict__ h, const float* __restrict__ W,  // [256,128]
    const float* __restrict__ bias, float* __restrict__ out) {
  const int lane = threadIdx.x & 31;
  const int t = blockIdx.y * 8 + (threadIdx.x >> 5);  // n-tile 0..15
  const int row0 = blockIdx.x * 16;
  const int lm = lane & 15;
  const int half = lane >> 4;
  const int n = t * 16 + lm;

  const float* arow = h + (size_t)(row0 + lm) * D + half * 2;
  const float* brow = W + (size_t)n * D + half * 2;
  v8f acc = {};
  for (int k = 0; k < D; k += 4) {
    const v2f a = *(const v2f*)(arow + k);
    const v2f b = *(const v2f*)(brow + k);
    acc = wmma_f32(a, b, acc);
  }
  const float bn = bias[n];
#pragma unroll
  for (int v = 0; v < 8; ++v) {
    const int row = row0 + v + half * 8;
    out[(size_t)row * OUT_STRIDE + D + n] = acc[v] + bn;
  }
}

// ---------------------------------------------------------------------------
extern "C" void kernel_launch(void* const* d_in, const int* in_sizes, int n_in,
                              void* d_out, int out_size, void* d_ws, size_t ws_size,
                              hipStream_t stream) {
  const float* x      = (const float*)d_in[0];
  const int*   ei     = (const int*)d_in[1];
  const float* pre_w  = (const float*)d_in[2];
  const float* pre_b  = (const float*)d_in[3];
  const float* pre_g  = (const float*)d_in[4];
  const float* pre_be = (const float*)d_in[5];
  const float* wl0    = (const float*)d_in[6];
  const float* wr0    = (const float*)d_in[7];
  const float* b0     = (const float*)d_in[8];
  const float* g0     = (const float*)d_in[9];
  const float* be0    = (const float*)d_in[10];
  const float* wl1    = (const float*)d_in[11];
  const float* wr1    = (const float*)d_in[12];
  const float* b1     = (const float*)d_in[13];
  const float* g1     = (const float*)d_in[14];
  const float* be1    = (const float*)d_in[15];
  const float* lin_w  = (const float*)d_in[16];
  const float* lin_b  = (const float*)d_in[17];
  float* out = (float*)d_out;

  // Workspace layout (floats): hA | hB | agg | cnt   (agg,cnt contiguous)
  float* ws  = (float*)d_ws;
  float* hA  = ws;
  float* hB  = ws + (size_t)N_NODES * D;
  float* agg = ws + 2 * (size_t)N_NODES * D;
  float* cnt = ws + 3 * (size_t)N_NODES * D;

  const int* src = ei;             // edge_index[0]
  const int* dst = ei + N_EDGES;   // edge_index[1]

  const int rowBlocks = (N_ROWTILES + 7) / 8;          // 8 waves/block
  const int zn = N_NODES * D + N_NODES;                // agg + cnt
  const int zBlocks = (zn + 255) / 256;
  const int eBlocks = (N_EDGES + 7) / 8;               // wave per edge

  // pre-MP: hA = ReLU(LN(x @ pre_w^T + pre_b))
  k_lin_ln_relu<<<rowBlocks, 256, 0, stream>>>(x, pre_w, pre_b, pre_g, pre_be, hA);

  // SAGE layer 0
  k_zero<<<zBlocks, 256, 0, stream>>>(agg, zn);
  k_aggregate<<<eBlocks, 256, 0, stream>>>(src, dst, hA, agg, cnt);
  k_sage_ln_relu<<<rowBlocks, 256, 0, stream>>>(agg, cnt, hA, wl0, wr0, b0, g0, be0, hB);

  // SAGE layer 1
  k_zero<<<zBlocks, 256, 0, stream>>>(agg, zn);
  k_aggregate<<<eBlocks, 256, 0, stream>>>(src, dst, hB, agg, cnt);
  k_sage_ln_relu<<<rowBlocks, 256, 0, stream>>>(agg, cnt, hB, wl1, wr1, b1, g1, be1, hA);

  // Output: concat([x, hA @ lin_w^T + lin_b])
  k_copy_x<<<(N_NODES * 32 + 255) / 256, 256, 0, stream>>>(x, out);
  dim3 fg(N_NODES / 16, 2);
  k_final_gemm<<<fg, 256, 0, stream>>>(hA, lin_w, lin_b, out);
}